// CompositeLoss_16054587752992
// MI455X (gfx1250) — compile-verified
//
#include <hip/hip_runtime.h>
#include <hip/hip_bf16.h>
#include <math.h>

typedef float v2f __attribute__((ext_vector_type(2)));
typedef float v8f __attribute__((ext_vector_type(8)));

#define LAMBDA_CHAMFER 1.0f
#define LAMBDA_EDGE    0.1f

// ws float layout: [0]=sum(min dist pred->gt), [1]=sum(min dist gt->pred),
//                  [2]=sum(edge len), [3]=sum(edge len^2)

__global__ void init_ws_kernel(float* ws) {
    if (threadIdx.x < 4) ws[threadIdx.x] = 0.0f;
}

// Directed chamfer pass: for each point p in P (NP per batch), min over q in Q
// (NQ per batch) of ||p-q||, summed (atomically) into *sum_out.
// Block = 256 threads (8 waves). Each wave owns one 16-row tile of P and
// sweeps all of Q in 16-column tiles, one V_WMMA_F32_16X16X4_F32 per tile.
// gridDim.x = NP/128, gridDim.y = B.
__global__ __launch_bounds__(256)
void chamfer_min_pass(const float* __restrict__ P, const float* __restrict__ Q,
                      int NP, int NQ, float* __restrict__ sum_out) {
    const int  lane = threadIdx.x & 31;
    const int  wave = threadIdx.x >> 5;
    const int  b    = blockIdx.y;
    const bool hi   = (lane & 16) != 0;  // lanes 16-31 hold K=2,3
    const int  l16  = lane & 15;

    const float* Pb = P + (size_t)b * NP * 3;
    const float* Qb = Q + (size_t)b * NQ * 3;

    const int prow = blockIdx.x * 128 + wave * 16;   // this wave's P tile base

    // ---- A tile: A[M][K] with M = l16; augmented coords (px,py,pz,1)
    const int   pi = prow + l16;
    const float px = Pb[pi * 3 + 0];
    const float py = Pb[pi * 3 + 1];
    const float pz = Pb[pi * 3 + 2];
    const float sq_self = px * px + py * py + pz * pz;
    v2f av;
    av.x = hi ? pz   : px;
    av.y = hi ? 1.0f : py;

    // sqp_local[v] = ||P[prow + v + hi*8]||^2  (D-tile row M = v + hi*8)
    float sqp_local[8];
#pragma unroll
    for (int v = 0; v < 8; ++v)
        sqp_local[v] = __shfl(sq_self, v + (hi ? 8 : 0), 32);

    v8f rmin, czero;
#pragma unroll
    for (int v = 0; v < 8; ++v) { rmin[v] = 3.0e38f; czero[v] = 0.0f; }

    for (int qbase = 0; qbase < NQ; qbase += 16) {
        // Branchless: every lane loads all 3 components of its Q point
        // (data is L2-resident; avoids exec-mask divergence in the hot loop).
        const float* qp = Qb + (size_t)(qbase + l16) * 3;
        const float qx = qp[0];
        const float qy = qp[1];
        const float qz = qp[2];
        const float sq = qx * qx + qy * qy + qz * qz;

        // B col (gt point n): (-2qx, -2qy, -2qz, ||q||^2); this lane supplies
        // K=0,1 (lanes 0-15) or K=2,3 (lanes 16-31) -> two v_cndmask selects.
        v2f bv;
        bv.x = -2.0f * (hi ? qz : qx);
        bv.y = hi ? sq : (-2.0f * qy);

        // D[M][N] = ||Q[N]||^2 - 2 * dot(P[M], Q[N])   (dist^2 minus ||P[M]||^2)
        v8f acc = __builtin_amdgcn_wmma_f32_16x16x4_f32(
            false, av, false, bv, (short)0, czero, false, false);

#pragma unroll
        for (int v = 0; v < 8; ++v) rmin[v] = fminf(rmin[v], acc[v]);
    }

    // Reduce the min across the 16 lanes of each half (N dimension).
#pragma unroll
    for (int v = 0; v < 8; ++v) {
        float r = rmin[v];
        r = fminf(r, __shfl_xor(r, 1, 32));
        r = fminf(r, __shfl_xor(r, 2, 32));
        r = fminf(r, __shfl_xor(r, 4, 32));
        r = fminf(r, __shfl_xor(r, 8, 32));
        rmin[v] = r;
    }

    // Per-row distance: sqrt(max(min + ||p||^2, 0)); sum rows of this half.
    float s = 0.0f;
#pragma unroll
    for (int v = 0; v < 8; ++v)
        s += sqrtf(fmaxf(rmin[v] + sqp_local[v], 0.0f));
    s += __shfl_xor(s, 16, 32);          // combine the two halves (rows 0-7 & 8-15)

    if (lane == 0) atomicAdd(sum_out, s);
}

// Edge length regularization: accumulate sum(len) and sum(len^2) over B*E edges.
__global__ __launch_bounds__(256)
void edge_len_pass(const float* __restrict__ P, const int* __restrict__ edges,
                   int B, int N, int E, float* __restrict__ ws) {
    const int lane  = threadIdx.x & 31;
    const int idx   = blockIdx.x * blockDim.x + threadIdx.x;
    const int total = B * E;

    float l = 0.0f, l2 = 0.0f;
    if (idx < total) {
        const int b = idx / E;
        const int e = idx - b * E;
        const int i0 = edges[2 * e + 0];
        const int i1 = edges[2 * e + 1];
        const float* Pb = P + (size_t)b * N * 3;
        const float dx = Pb[i0 * 3 + 0] - Pb[i1 * 3 + 0];
        const float dy = Pb[i0 * 3 + 1] - Pb[i1 * 3 + 1];
        const float dz = Pb[i0 * 3 + 2] - Pb[i1 * 3 + 2];
        l  = sqrtf(dx * dx + dy * dy + dz * dz);
        l2 = l * l;
    }
#pragma unroll
    for (int off = 1; off < 32; off <<= 1) {
        l  += __shfl_xor(l,  off, 32);
        l2 += __shfl_xor(l2, off, 32);
    }
    if (lane == 0) {
        atomicAdd(&ws[2], l);
        atomicAdd(&ws[3], l2);
    }
}

__global__ void finalize_kernel(const float* __restrict__ ws, float* __restrict__ out,
                                int BN, int BM, int NE) {
    const float cd  = ws[0] / (float)BN + ws[1] / (float)BM;
    const float n   = (float)NE;
    const float var = (ws[3] - ws[2] * ws[2] / n) / (n - 1.0f);  // ddof=1
    out[0] = LAMBDA_CHAMFER * cd + LAMBDA_EDGE * var;
}

extern "C" void kernel_launch(void* const* d_in, const int* in_sizes, int n_in,
                              void* d_out, int out_size, void* d_ws, size_t ws_size,
                              hipStream_t stream) {
    const float* pred  = (const float*)d_in[0];   // (B, N, 3) f32
    const float* gt    = (const float*)d_in[1];   // (B, M, 3) f32
    const int*   edges = (const int*)  d_in[2];   // (E, 2)

    const int B = 8;
    const int N = in_sizes[0] / (3 * B);          // 4096
    const int M = in_sizes[1] / (3 * B);          // 4096
    const int E = in_sizes[2] / 2;                // 12288

    float* ws  = (float*)d_ws;
    float* out = (float*)d_out;

    init_ws_kernel<<<1, 32, 0, stream>>>(ws);

    // pred -> gt direction (per-pred mins), then gt -> pred direction.
    chamfer_min_pass<<<dim3(N / 128, B), 256, 0, stream>>>(pred, gt, N, M, &ws[0]);
    chamfer_min_pass<<<dim3(M / 128, B), 256, 0, stream>>>(gt, pred, M, N, &ws[1]);

    const int totalE = B * E;
    edge_len_pass<<<(totalE + 255) / 256, 256, 0, stream>>>(pred, edges, B, N, E, ws);

    finalize_kernel<<<1, 1, 0, stream>>>(ws, out, B * N, B * M, totalE);
}